// CudaLaplacian1DLinearAttention_11742440588159
// MI455X (gfx1250) — compile-verified
//
#include <hip/hip_runtime.h>

// ---------------------------------------------------------------------------
// CDNA5 (gfx1250, wave32) Laplacian-kernel attention, fp32 WMMA path
// + async LDS double-buffering (GLOBAL_LOAD_ASYNC_TO_LDS_B128 / ASYNCcnt).
// ---------------------------------------------------------------------------

typedef float v2f __attribute__((ext_vector_type(2)));
typedef float v8f __attribute__((ext_vector_type(8)));

#define DIM      1024
#define HEADS    16
#define HEAD_DIM 64
#define BATCH    2
#define SEQ      1024
#define LAMBDA_  4.0f
#define EPS_     1e-6f

// ---------------------------------------------------------------------------
// fp32 WMMA GEMM:  out[m][n] = sum_k A[m*K+k] * W[n*K+k] + bias[n]
// One wave computes a 32x32 output tile (4 C fragments) so each A/B fragment
// is reused twice: per K-step 4 b64 loads feed 4 v_wmma_f32_16x16x4_f32.
// Block = 128 threads = 4 waves -> 32x128 strip of the output.
// ---------------------------------------------------------------------------
__global__ __launch_bounds__(128)
void wmma_gemm_nt(const float* __restrict__ A, const float* __restrict__ W,
                  const float* __restrict__ bias, float* __restrict__ out,
                  int N, int K) {
    const int lane = threadIdx.x & 31;
    const int wave = threadIdx.x >> 5;
    const int m0   = blockIdx.y * 32;
    const int n0   = blockIdx.x * 128 + wave * 32;
    const int mn   = lane & 15;   // A-frag row / B-frag col fed by this lane
    const int hi   = lane >> 4;   // K-pair select {0,1} vs {2,3}

    // A-frag layout (ISA 7.12.2, 32-bit A 16x4): lane m holds A[m][2*hi + {0,1}]
    const float* arow0 = A + (size_t)(m0 + mn) * K + 2 * hi;
    const float* arow1 = arow0 + (size_t)16 * K;
    // B[k][n] = W[n][k]; VGPR r, half-wave hi -> B[k0 + 2*hi + r][n]
    const float* wrow0 = W + (size_t)(n0 + mn) * K + 2 * hi;
    const float* wrow1 = wrow0 + (size_t)16 * K;

    v8f c00 = {}, c01 = {}, c10 = {}, c11 = {};
    for (int k0 = 0; k0 < K; k0 += 4) {
        v2f a0 = *(const v2f*)(arow0 + k0);
        v2f a1 = *(const v2f*)(arow1 + k0);
        v2f b0 = *(const v2f*)(wrow0 + k0);
        v2f b1 = *(const v2f*)(wrow1 + k0);
        c00 = __builtin_amdgcn_wmma_f32_16x16x4_f32(false, a0, false, b0, (short)0, c00, false, false);
        c01 = __builtin_amdgcn_wmma_f32_16x16x4_f32(false, a0, false, b1, (short)0, c01, false, false);
        c10 = __builtin_amdgcn_wmma_f32_16x16x4_f32(false, a1, false, b0, (short)0, c10, false, false);
        c11 = __builtin_amdgcn_wmma_f32_16x16x4_f32(false, a1, false, b1, (short)0, c11, false, false);
    }

    // C/D layout: VGPR r -> row (r + 8*hi), col = lane&15
    const float bn0 = bias[n0 + mn];
    const float bn1 = bias[n0 + 16 + mn];
#pragma unroll
    for (int r = 0; r < 8; ++r) {
        const int row0 = m0 + r + 8 * hi;
        const int row1 = row0 + 16;
        out[(size_t)row0 * N + (n0 + mn)]      = c00[r] + bn0;
        out[(size_t)row0 * N + (n0 + 16 + mn)] = c01[r] + bn1;
        out[(size_t)row1 * N + (n0 + mn)]      = c10[r] + bn0;
        out[(size_t)row1 * N + (n0 + 16 + mn)] = c11[r] + bn1;
    }
}

// ---------------------------------------------------------------------------
// Async-copy one 16x64 f32 tile (k and v) into LDS: 4 async b128 per wave.
// GVS mode: mem = SGPR base + 32-bit VGPR byte offset. Tracked by ASYNCcnt.
// ---------------------------------------------------------------------------
__device__ __forceinline__
void async_tile_issue(const float* __restrict__ krow, const float* __restrict__ vrow,
                      float4* kdst, float4* vdst, int tid) {
#pragma unroll
    for (int e = tid; e < 256; e += 128) {        // 16 rows x 16 float4
        const int r = e >> 4, c4 = e & 15;
        const unsigned goff = (unsigned)(r * (3 * DIM * 4) + c4 * 16); // bytes
        const unsigned kl = (unsigned)(uintptr_t)(kdst + e);           // LDS byte addr
        const unsigned vl = (unsigned)(uintptr_t)(vdst + e);
        asm volatile("global_load_async_to_lds_b128 %0, %1, %2"
                     :: "v"(kl), "v"(goff), "s"(krow) : "memory");
        asm volatile("global_load_async_to_lds_b128 %0, %1, %2"
                     :: "v"(vl), "v"(goff), "s"(vrow) : "memory");
    }
}

// ---------------------------------------------------------------------------
// Laplacian (L1) kernel attention.  qkv buffer: [B*S, 3*DIM] row-major,
// q|k|v split along columns, head h -> columns h*64..h*64+63.
// Block = 128 thr = 4 waves; each wave owns 16 query rows (q row in VGPRs).
// Per 16-key tile: each lane computes exactly its 8 A-fragment kernel
// entries (VALU |q-k| + v_exp_f32), then 16 WMMAs accumulate kern @ v.
// k/v tiles double-buffered in LDS via async loads.
// ---------------------------------------------------------------------------
__global__ __launch_bounds__(128)
void lap_attn(const float* __restrict__ qkv, float* __restrict__ y) {
    __shared__ float4 kv4[2][2][16 * 16];  // [buffer][k|v][16 rows x 16 float4]
    __shared__ float  rs_[4][16];          // per-wave row sums

    const int tid  = threadIdx.x;
    const int lane = tid & 31;
    const int wave = tid >> 5;
    const int bh   = blockIdx.y;
    const int b    = bh >> 4;              // / HEADS
    const int h    = bh & 15;
    const int ib   = blockIdx.x * 64 + wave * 16;  // wave's query base row
    const int m    = lane & 15;
    const int hi   = lane >> 4;

    const size_t rstride = 3 * DIM;
    const float* qbase = qkv + (size_t)(b * SEQ) * rstride + h * HEAD_DIM;
    const float* kbase = qbase + DIM;
    const float* vbase = qbase + 2 * DIM;

    // q row (ib+m): 64 f32 = 16 float4 in VGPRs (lanes m and m+16 duplicate)
    float4 q[16];
    {
        const float4* qp = (const float4*)(qbase + (size_t)(ib + m) * rstride);
#pragma unroll
        for (int i = 0; i < 16; ++i) q[i] = qp[i];
    }

    v8f acc[4] = {};          // 16 rows x 64 dh as four 16x16 f32 C fragments
    float rowsum = 0.0f;

    // prologue: async-load tile 0 into buffer 0
    async_tile_issue(kbase, vbase, &kv4[0][0][0], &kv4[0][1][0], tid);

    int buf = 0;
    for (int jb = 0; jb < SEQ; jb += 16) {
        __syncthreads();   // all waves done reading buf^1 (previous tile)
        if (jb + 16 < SEQ) {
            async_tile_issue(kbase + (size_t)(jb + 16) * rstride,
                             vbase + (size_t)(jb + 16) * rstride,
                             &kv4[buf ^ 1][0][0], &kv4[buf ^ 1][1][0], tid);
            // async loads complete in order: <=4 outstanding means the
            // 4 for the *current* tile have landed; next tile stays in flight
            asm volatile("s_wait_asynccnt 4" ::: "memory");
        } else {
            asm volatile("s_wait_asynccnt 0" ::: "memory");
        }
        __syncthreads();   // every wave's portion of current tile visible

        const float4* ks4 = &kv4[buf][0][0];
        const float*  vsf = (const float*)&kv4[buf][1][0];

        // 8 kernel entries per lane -> A fragments for the 4 K-steps
        v2f afrag[4];
#pragma unroll
        for (int p = 0; p < 4; ++p) {
#pragma unroll
            for (int half = 0; half < 2; ++half) {
                const int j = 4 * p + 2 * hi + half;       // key within tile
                const float4* kr = &ks4[j * 16];
                float dist = 0.0f;
#pragma unroll
                for (int d4 = 0; d4 < 16; ++d4) {
                    float4 kv = kr[d4];
                    dist += __builtin_fabsf(q[d4].x - kv.x)
                          + __builtin_fabsf(q[d4].y - kv.y)
                          + __builtin_fabsf(q[d4].z - kv.z)
                          + __builtin_fabsf(q[d4].w - kv.w);
                }
                float kern = __expf(dist * (-LAMBDA_ / (float)HEAD_DIM));
                rowsum += kern;
                if (half == 0) afrag[p].x = kern; else afrag[p].y = kern;
            }
        }

        // kern(16x16) @ v(16x64): 4 N-subtiles x 4 K-steps of 16x16x4 WMMA
#pragma unroll
        for (int t = 0; t < 4; ++t) {
#pragma unroll
            for (int p = 0; p < 4; ++p) {
                const int krow = 4 * p + 2 * hi;
                v2f bfrag;
                bfrag.x = vsf[(krow    ) * 64 + t * 16 + m];
                bfrag.y = vsf[(krow + 1) * 64 + t * 16 + m];
                acc[t] = __builtin_amdgcn_wmma_f32_16x16x4_f32(
                             false, afrag[p], false, bfrag,
                             (short)0, acc[t], false, false);
            }
        }
        buf ^= 1;
    }

    // total row sum: lanes m and m+16 hold halves of row m (wave32 shuffle)
    rowsum += __shfl_xor(rowsum, 16, 32);
    if (hi == 0) rs_[wave][m] = rowsum;   // intra-wave DS ordering suffices

    // normalize + store: row r+8*hi, col t*16+m -> y[b][s][h*64+d]
#pragma unroll
    for (int r = 0; r < 8; ++r) {
        const int row = r + 8 * hi;
        const float inv = 1.0f / (rs_[wave][row] + EPS_);
        const int s = ib + row;
        float* yr = y + (size_t)(b * SEQ + s) * DIM + h * HEAD_DIM + m;
#pragma unroll
        for (int t = 0; t < 4; ++t)
            yr[t * 16] = acc[t][r] * inv;
    }
}

// ---------------------------------------------------------------------------
// Depthwise conv1d (kernel 3, pad 1, cross-correlation per XLA) on v,
// accumulated into y:  y[b][s][c] += b[c] + sum_r v[b][s-1+r][c] * w[c][r]
// ---------------------------------------------------------------------------
__global__ __launch_bounds__(256)
void dwconv_add(const float* __restrict__ qkv, const float* __restrict__ w,
                const float* __restrict__ bias, float* __restrict__ y) {
    const int idx = blockIdx.x * 256 + threadIdx.x;  // over B*S*DIM (exact)
    const int c = idx & (DIM - 1);
    const int s = (idx >> 10) & (SEQ - 1);
    const int b = idx >> 20;
    const int h = c >> 6, d = c & 63;

    const float* vb = qkv + (size_t)(b * SEQ) * (3 * DIM) + 2 * DIM + h * HEAD_DIM + d;
    float acc = bias[c];
    if (s > 0)       acc += vb[(size_t)(s - 1) * (3 * DIM)] * w[c * 3 + 0];
    acc                  += vb[(size_t)(s    ) * (3 * DIM)] * w[c * 3 + 1];
    if (s < SEQ - 1) acc += vb[(size_t)(s + 1) * (3 * DIM)] * w[c * 3 + 2];
    y[idx] += acc;
}

// ---------------------------------------------------------------------------
extern "C" void kernel_launch(void* const* d_in, const int* in_sizes, int n_in,
                              void* d_out, int out_size, void* d_ws, size_t ws_size,
                              hipStream_t stream) {
    const float* x      = (const float*)d_in[0];   // [B,S,DIM]
    const float* qkv_w  = (const float*)d_in[1];   // [3*DIM, DIM]
    const float* qkv_b  = (const float*)d_in[2];   // [3*DIM]
    const float* proj_w = (const float*)d_in[3];   // [DIM, DIM]
    const float* proj_b = (const float*)d_in[4];   // [DIM]
    const float* dwc_w  = (const float*)d_in[5];   // [DIM,1,3]
    const float* dwc_b  = (const float*)d_in[6];   // [DIM]
    float* out = (float*)d_out;                    // [B,S,DIM]

    const int M = BATCH * SEQ;                     // 2048
    float* qkv_buf = (float*)d_ws;                               // M x 3*DIM (24 MB)
    float* y_buf   = qkv_buf + (size_t)M * 3 * DIM;              // M x DIM   (8 MB)

    // 1) QKV projection: [2048,1024] @ [1024,3072(T)] -> [2048,3072]
    wmma_gemm_nt<<<dim3((3 * DIM) / 128, M / 32), 128, 0, stream>>>(
        x, qkv_w, qkv_b, qkv_buf, 3 * DIM, DIM);

    // 2) Laplacian-kernel attention -> y_buf (fully overwritten)
    lap_attn<<<dim3(SEQ / 64, BATCH * HEADS), 128, 0, stream>>>(qkv_buf, y_buf);

    // 3) local depthwise-conv branch accumulated into y_buf
    dwconv_add<<<(M * DIM) / 256, 256, 0, stream>>>(qkv_buf, dwc_w, dwc_b, y_buf);

    // 4) output projection: [2048,1024] @ [1024,1024(T)] + b -> out
    wmma_gemm_nt<<<dim3(DIM / 128, M / 32), 128, 0, stream>>>(
        y_buf, proj_w, proj_b, out, DIM, DIM);
}